// PR2d_56289841381419
// MI455X (gfx1250) — compile-verified
//
#include <hip/hip_runtime.h>
#include <math.h>

// ---------------------------------------------------------------------------
// PR2d pole-residue spectral operator for MI455X (gfx1250, wave32, WMMA).
// Heavy stages are f32 complex GEMMs on V_WMMA_F32_16X16X4_F32, with
// branchless (cndmask) padding and Gauss 3-multiply complex products.
// ---------------------------------------------------------------------------

#define TL1 41   // theta grid
#define TL2 40   // z grid
#define NB  64   // batch
#define CI  32   // in channels
#define CO  32   // out channels
#define M1  8
#define M2  8
#define PADK 48  // padded DFT matrix dim (multiple of 16 >= 41)

static __device__ __forceinline__ float fPI() { return 3.14159265358979323846f; }

typedef __attribute__((ext_vector_type(2))) float v2f;
typedef __attribute__((ext_vector_type(8))) float v8f;

// D = A(16x4) * B(4x16) + C  in f32 on the matrix pipe.
static __device__ __forceinline__ v8f wmma4(v2f a, v2f b, v8f c) {
  return __builtin_amdgcn_wmma_f32_16x16x4_f32(
      false, a, false, b, (short)0, c, false, false);
}

// Unchecked A fragment: 16x4 tile, the per-lane K pair is row-contiguous ->
// one 64-bit load.  Caller guarantees in-bounds.
static __device__ __forceinline__ v2f ldA_nc(const float* A, int lda,
                                             int mbase, int k0) {
  int lane = threadIdx.x & 31;
  int m = mbase + (lane & 15);
  int kk = k0 + 2 * (lane >> 4);
  float2 t = *reinterpret_cast<const float2*>(A + m * lda + kk);
  v2f a; a.x = t.x; a.y = t.y;
  return a;
}

// Unchecked B fragment: 4x16 tile (K rows x N cols).  Caller guarantees
// in-bounds (use only on padded/exact operands).
static __device__ __forceinline__ v2f ldB_nc(const float* Bm, int ldb, int k0,
                                             int nbase) {
  int lane = threadIdx.x & 31;
  int n = nbase + (lane & 15);
  int kk = k0 + 2 * (lane >> 4);
  v2f b;
  b.x = Bm[kk * ldb + n];
  b.y = Bm[(kk + 1) * ldb + n];
  return b;
}

// Branchless masked B fragment for a 41x40 slab: clamp index into the valid
// region and cndmask the value -- no EXEC-mask branching, plain loads.
static __device__ __forceinline__ v2f ldB_slab(const float* Bm, int k0,
                                               int nbase) {
  int lane = threadIdx.x & 31;
  int n = nbase + (lane & 15);
  int kk = k0 + 2 * (lane >> 4);
  bool nv = (n < TL2);
  bool v0 = (kk < TL1) & nv;
  bool v1 = (kk + 1 < TL1) & nv;
  int i0 = kk * TL2 + n;
  int i1 = i0 + TL2;
  float x0 = Bm[v0 ? i0 : 0];
  float x1 = Bm[v1 ? i1 : 0];
  v2f b;
  b.x = v0 ? x0 : 0.0f;
  b.y = v1 ? x1 : 0.0f;
  return b;
}

// C tile store: vgpr r, lanes 0-15 -> M=r, lanes 16-31 -> M=8+r; N=lane%16.
static __device__ __forceinline__ void stC(float* C, int ldc, int mbase,
                                           int nbase, v8f c, int nMax) {
  int lane = threadIdx.x & 31;
  int n = nbase + (lane & 15);
  int mo = mbase + 8 * (lane >> 4);
  if (n < nMax) {
#pragma unroll
    for (int r = 0; r < 8; ++r) C[(mo + r) * ldc + n] = c[r];
  }
}

// ---------------------------------------------------------------------------
// Kernel 1: DFT / iDFT matrices, zero-padded to 48x48.
// ---------------------------------------------------------------------------
__global__ __launch_bounds__(256) void k_const(float* WZC, float* WZS,
                                               float* WTRe, float* WTIm,
                                               float* WITRe, float* WITIm,
                                               float* WIZC, float* WIZS) {
  for (int idx = threadIdx.x; idx < PADK * PADK; idx += blockDim.x) {
    int r = idx / PADK, c = idx % PADK;
    float vzc = 0.f, vzs = 0.f, wtr = 0.f, wti = 0.f;
    float witr = 0.f, witi = 0.f, wizc = 0.f, wizs = 0.f;
    if (r < TL2 && c < TL2) {
      float ang = 2.0f * fPI() * (float)(r * c) / (float)TL2;
      vzc = cosf(ang); vzs = -sinf(ang);     // forward z-DFT (e^{-i})
      wizc = cosf(ang); wizs = sinf(ang);    // inverse z-DFT (e^{+i})
    }
    if (r < TL1 && c < TL1) {
      float ang = 2.0f * fPI() * (float)(r * c) / (float)TL1;
      wtr = cosf(ang); wti = -sinf(ang);     // forward theta-DFT
      witr = cosf(ang); witi = sinf(ang);    // inverse theta-DFT
    }
    WZC[idx] = vzc;  WZS[idx] = vzs;
    WTRe[idx] = wtr; WTIm[idx] = wti;
    WITRe[idx] = witr; WITIm[idx] = witi;
    WIZC[idx] = wizc;  WIZS[idx] = wizs;
  }
}

// ---------------------------------------------------------------------------
// Kernel 2: per-(i,k) weight tables: r1, r2, e1, e2, freq-major H.
// ---------------------------------------------------------------------------
__global__ __launch_bounds__(256) void k_weights(
    const float* w1, const float* w2, const float* wr, float* r1re,
    float* r1im, float* r2re, float* r2im, float* e1re, float* e1im,
    float* e2re, float* e2im, float* HfRe, float* HfIm) {
  __shared__ float s_p1[M1 * 2], s_p2[M2 * 2], s_res[M1 * M2 * 2];
  __shared__ float s_r1[TL1 * M1 * 2];
  __shared__ float s_r2[TL2 * M2 * 2];
  __shared__ float s_tmp[M1 * TL2 * 2];
  int ik = blockIdx.x;  // i*CO + k
  int tid = threadIdx.x;
  if (tid < M1 * 2) s_p1[tid] = w1[ik * M1 * 2 + tid];
  if (tid < M2 * 2) s_p2[tid] = w2[ik * M2 * 2 + tid];
  if (tid < M1 * M2 * 2) s_res[tid] = wr[ik * M1 * M2 * 2 + tid];
  __syncthreads();
  // r1[o][p] = 1/(i*mu1(o) - p1[p]);  mu1 = (40/41)*signed index
  for (int t = tid; t < TL1 * M1; t += blockDim.x) {
    int o = t / M1, p = t % M1;
    int fo = (o <= 20) ? o : o - TL1;
    float mu = (40.0f / 41.0f) * (float)fo;
    float dre = -s_p1[2 * p], dim = mu - s_p1[2 * p + 1];
    float inv = 1.0f / (dre * dre + dim * dim);
    float rr = dre * inv, ri = -dim * inv;
    s_r1[2 * t] = rr; s_r1[2 * t + 1] = ri;
    r1re[(ik * TL1 + o) * M1 + p] = rr;
    r1im[(ik * TL1 + o) * M1 + p] = ri;
  }
  // r2[x][q] = 1/(i*mu2(x) - p2[q]);  mu2 = 2*pi*39/600 * signed index
  for (int t = tid; t < TL2 * M2; t += blockDim.x) {
    int x = t / M2, q = t % M2;
    int fx = (x < 20) ? x : x - TL2;
    float mu = (2.0f * fPI() * 39.0f / 600.0f) * (float)fx;
    float dre = -s_p2[2 * q], dim = mu - s_p2[2 * q + 1];
    float inv = 1.0f / (dre * dre + dim * dim);
    float rr = dre * inv, ri = -dim * inv;
    s_r2[2 * t] = rr; s_r2[2 * t + 1] = ri;
    r2re[(ik * TL2 + x) * M2 + q] = rr;
    r2im[(ik * TL2 + x) * M2 + q] = ri;
  }
  // e1[p][t] = exp(p1[p] * theta_t), theta step = 2pi/40
  for (int t = tid; t < M1 * TL1; t += blockDim.x) {
    int p = t / TL1, tt = t % TL1;
    float th = (2.0f * fPI() / 40.0f) * (float)tt;
    float a = s_p1[2 * p], bb = s_p1[2 * p + 1];
    float ea = expf(a * th);
    e1re[(ik * M1 + p) * TL1 + tt] = ea * cosf(bb * th);
    e1im[(ik * M1 + p) * TL1 + tt] = ea * sinf(bb * th);
  }
  // e2[q][x] = exp(p2[q] * z_x), z step = 15/39
  for (int t = tid; t < M2 * TL2; t += blockDim.x) {
    int q = t / TL2, x = t % TL2;
    float z = (15.0f / 39.0f) * (float)x;
    float a = s_p2[2 * q], bb = s_p2[2 * q + 1];
    float ea = expf(a * z);
    e2re[(ik * M2 + q) * TL2 + x] = ea * cosf(bb * z);
    e2im[(ik * M2 + q) * TL2 + x] = ea * sinf(bb * z);
  }
  __syncthreads();
  // tmp[p][x] = sum_q res[p][q] * r2[x][q]
  for (int t = tid; t < M1 * TL2; t += blockDim.x) {
    int p = t / TL2, x = t % TL2;
    float ar = 0.f, ai = 0.f;
    for (int q = 0; q < M2; ++q) {
      float rr = s_res[(p * M2 + q) * 2], riq = s_res[(p * M2 + q) * 2 + 1];
      float br = s_r2[(x * M2 + q) * 2], bi = s_r2[(x * M2 + q) * 2 + 1];
      ar += rr * br - riq * bi;
      ai += rr * bi + riq * br;
    }
    s_tmp[2 * t] = ar; s_tmp[2 * t + 1] = ai;
  }
  __syncthreads();
  // Hf[o][x] = sum_p r1[o][p]*tmp[p][x]   (freq-major store)
  for (int t = tid; t < TL1 * TL2; t += blockDim.x) {
    int o = t / TL2, x = t % TL2;
    float hr = 0.f, hi = 0.f;
    for (int p = 0; p < M1; ++p) {
      float ar = s_r1[(o * M1 + p) * 2], ai = s_r1[(o * M1 + p) * 2 + 1];
      float br = s_tmp[(p * TL2 + x) * 2], bi = s_tmp[(p * TL2 + x) * 2 + 1];
      hr += ar * br - ai * bi;
      hi += ar * bi + ai * br;
    }
    HfRe[t * (CI * CO) + ik] = hr;
    HfIm[t * (CI * CO) + ik] = hi;
  }
}

// ---------------------------------------------------------------------------
// Kernel 3: forward z-DFT.  alphaZ[m][x'] = sum_x X[m][x]*W[x][x'].
// M = B*CI*L1 = 83968 real rows.  All loads in-bounds by construction.
// ---------------------------------------------------------------------------
__global__ __launch_bounds__(256) void k_zdft(const float* X, const float* WZC,
                                              const float* WZS, float* aZre,
                                              float* aZim) {
  int wave = blockIdx.x * (blockDim.x >> 5) + (threadIdx.x >> 5);
  int mt = wave / 3, nt = wave % 3;
  int mbase = mt * 16, nbase = nt * 16;
  v8f cre = {}; v8f cim = {};
#pragma unroll
  for (int k0 = 0; k0 < TL2; k0 += 4) {
    v2f a  = ldA_nc(X, TL2, mbase, k0);       // kk <= 39 < 40
    v2f bc = ldB_nc(WZC, PADK, k0, nbase);    // 48x48 padded
    v2f bs = ldB_nc(WZS, PADK, k0, nbase);
    cre = wmma4(a, bc, cre);
    cim = wmma4(a, bs, cim);
  }
  stC(aZre, TL2, mbase, nbase, cre, TL2);
  stC(aZim, TL2, mbase, nbase, cim, TL2);
}

// ---------------------------------------------------------------------------
// Kernel 4: forward theta-DFT, batched over 2048 (b,i) slabs:
// alpha[o'][x] = sum_o E[o'][o] * in[o][x].  Gauss 3-mult complex GEMM.
// Dual store: (b,i,o,x) planes + freq-major aF[(o*L2+x)*2048 + slab].
// ---------------------------------------------------------------------------
__global__ __launch_bounds__(256) void k_tdft(const float* aZre,
                                              const float* aZim,
                                              const float* WTRe,
                                              const float* WTIm, float* aRe,
                                              float* aIm, float* aFre,
                                              float* aFim) {
  int wave = blockIdx.x * (blockDim.x >> 5) + (threadIdx.x >> 5);
  int slab = wave / 9;
  int t9 = wave % 9;
  int mt = t9 / 3, nt = t9 % 3;
  int mbase = mt * 16, nbase = nt * 16;
  const float* inre = aZre + slab * TL1 * TL2;
  const float* inim = aZim + slab * TL1 * TL2;
  v8f p1 = {}; v8f p2 = {}; v8f p3 = {};
#pragma unroll
  for (int k0 = 0; k0 < 44; k0 += 4) {
    v2f ar = ldA_nc(WTRe, PADK, mbase, k0);   // 48x48 padded
    v2f ai = ldA_nc(WTIm, PADK, mbase, k0);
    v2f br = ldB_slab(inre, k0, nbase);       // branchless masked
    v2f bi = ldB_slab(inim, k0, nbase);
    v2f af = ar + ai;
    v2f bf = br + bi;
    p1 = wmma4(ar, br, p1);
    p2 = wmma4(ai, bi, p2);
    p3 = wmma4(af, bf, p3);
  }
  int lane = threadIdx.x & 31;
  int n = nbase + (lane & 15);
  int mo = mbase + 8 * (lane >> 4);
  if (n < TL2) {
#pragma unroll
    for (int r = 0; r < 8; ++r) {
      int m = mo + r;
      if (m < TL1) {
        float vr = p1[r] - p2[r];
        float vi = p3[r] - p1[r] - p2[r];
        int idx = (slab * TL1 + m) * TL2 + n;
        aRe[idx] = vr; aIm[idx] = vi;
        int fidx = (m * TL2 + n) * (NB * CI) + slab;
        aFre[fidx] = vr; aFim[fidx] = vi;
      }
    }
  }
}

// ---------------------------------------------------------------------------
// Kernel 5: out_res2[b,k,p,q] = sum_i res[i,k,p,q] * sum_x (sum_o
// r1[o,p]*alpha[b,i,o,x]) * r2[x,q].  Block per (b,k), LDS staged.
// ---------------------------------------------------------------------------
__global__ __launch_bounds__(256) void k_S(const float* aRe, const float* aIm,
                                           const float* r1re,
                                           const float* r1im,
                                           const float* r2re,
                                           const float* r2im, const float* wr,
                                           float* res2re, float* res2im) {
  int bk = blockIdx.x;
  int b = bk / CO, k = bk % CO;
  int tid = threadIdx.x;
  __shared__ float s_a[TL1 * TL2 * 2];
  __shared__ float s_r1[TL1 * M1 * 2];
  __shared__ float s_r2[TL2 * M2 * 2];
  __shared__ float s_res[M1 * M2 * 2];
  __shared__ float s_T1[M1 * TL2 * 2];
  float accR = 0.f, accI = 0.f;
  for (int i = 0; i < CI; ++i) {
    int ik = i * CO + k;
    const float* ar = aRe + (b * CI + i) * TL1 * TL2;
    const float* ai = aIm + (b * CI + i) * TL1 * TL2;
    for (int t = tid; t < TL1 * TL2; t += blockDim.x) {
      s_a[2 * t] = ar[t]; s_a[2 * t + 1] = ai[t];
    }
    for (int t = tid; t < TL1 * M1; t += blockDim.x) {
      s_r1[2 * t] = r1re[ik * TL1 * M1 + t];
      s_r1[2 * t + 1] = r1im[ik * TL1 * M1 + t];
    }
    for (int t = tid; t < TL2 * M2; t += blockDim.x) {
      s_r2[2 * t] = r2re[ik * TL2 * M2 + t];
      s_r2[2 * t + 1] = r2im[ik * TL2 * M2 + t];
    }
    for (int t = tid; t < M1 * M2 * 2; t += blockDim.x)
      s_res[t] = wr[ik * M1 * M2 * 2 + t];
    __syncthreads();
    // T1[p][x] = sum_o r1[o][p] * alpha[o][x]
    for (int t = tid; t < M1 * TL2; t += blockDim.x) {
      int p = t / TL2, x = t % TL2;
      float tr = 0.f, ti = 0.f;
      for (int o = 0; o < TL1; ++o) {
        float rr = s_r1[(o * M1 + p) * 2], ri = s_r1[(o * M1 + p) * 2 + 1];
        float arr = s_a[(o * TL2 + x) * 2], aii = s_a[(o * TL2 + x) * 2 + 1];
        tr += rr * arr - ri * aii;
        ti += rr * aii + ri * arr;
      }
      s_T1[2 * t] = tr; s_T1[2 * t + 1] = ti;
    }
    __syncthreads();
    if (tid < M1 * M2) {
      int p = tid / M2, q = tid % M2;
      float sr = 0.f, si = 0.f;
      for (int x = 0; x < TL2; ++x) {
        float tr = s_T1[(p * TL2 + x) * 2], ti = s_T1[(p * TL2 + x) * 2 + 1];
        float rr = s_r2[(x * M2 + q) * 2], ri = s_r2[(x * M2 + q) * 2 + 1];
        sr += tr * rr - ti * ri;
        si += tr * ri + ti * rr;
      }
      float wrr = s_res[tid * 2], wri = s_res[tid * 2 + 1];
      accR += wrr * sr - wri * si;
      accI += wrr * si + wri * sr;
    }
    __syncthreads();
  }
  if (tid < M1 * M2) {
    res2re[bk * M1 * M2 + tid] = accR;
    res2im[bk * M1 * M2 + tid] = accI;
  }
}

// ---------------------------------------------------------------------------
// Kernel 6: frequency-pointwise channel mixing: for each of 1640 freqs,
// C(64x32) = A(64x32) * H(32x32) complex (Gauss 3-mult), scattered to
// (b,k,o,x).  One block = one frequency, 8 waves = 4x2 C tiles.
// All dims multiples of 16 -> fully unchecked loads/stores.
// ---------------------------------------------------------------------------
__global__ __launch_bounds__(256) void k_mix(const float* aFre,
                                             const float* aFim,
                                             const float* HfRe,
                                             const float* HfIm, float* R1re,
                                             float* R1im) {
  int f = blockIdx.x;  // o*L2 + x
  int w = threadIdx.x >> 5;
  int mt = w >> 1, nt = w & 1;
  const float* Ar = aFre + f * (NB * CI);
  const float* Ai = aFim + f * (NB * CI);
  const float* Br = HfRe + f * (CI * CO);
  const float* Bi = HfIm + f * (CI * CO);
  v8f p1 = {}; v8f p2 = {}; v8f p3 = {};
  int mbase = mt * 16, nbase = nt * 16;
#pragma unroll
  for (int k0 = 0; k0 < CI; k0 += 4) {
    v2f ar = ldA_nc(Ar, CI, mbase, k0);
    v2f ai = ldA_nc(Ai, CI, mbase, k0);
    v2f br = ldB_nc(Br, CO, k0, nbase);
    v2f bi = ldB_nc(Bi, CO, k0, nbase);
    v2f af = ar + ai;
    v2f bf = br + bi;
    p1 = wmma4(ar, br, p1);
    p2 = wmma4(ai, bi, p2);
    p3 = wmma4(af, bf, p3);
  }
  int lane = threadIdx.x & 31;
  int n = nbase + (lane & 15);        // k (out channel)
  int mo = mbase + 8 * (lane >> 4);   // b
#pragma unroll
  for (int r = 0; r < 8; ++r) {
    int bb = mo + r;
    int idx = (bb * CO + n) * (TL1 * TL2) + f;
    R1re[idx] = p1[r] - p2[r];
    R1im[idx] = p3[r] - p1[r] - p2[r];
  }
}

// ---------------------------------------------------------------------------
// Kernel 7: inverse theta-DFT (complex, Gauss 3-mult), 2048 (b,k) slabs.
// ---------------------------------------------------------------------------
__global__ __launch_bounds__(256) void k_itdft(const float* R1re,
                                               const float* R1im,
                                               const float* WITRe,
                                               const float* WITIm, float* Vre,
                                               float* Vim) {
  int wave = blockIdx.x * (blockDim.x >> 5) + (threadIdx.x >> 5);
  int slab = wave / 9;
  int t9 = wave % 9;
  int mt = t9 / 3, nt = t9 % 3;
  int mbase = mt * 16, nbase = nt * 16;
  const float* inre = R1re + slab * TL1 * TL2;
  const float* inim = R1im + slab * TL1 * TL2;
  v8f p1 = {}; v8f p2 = {}; v8f p3 = {};
#pragma unroll
  for (int k0 = 0; k0 < 44; k0 += 4) {
    v2f ar = ldA_nc(WITRe, PADK, mbase, k0);
    v2f ai = ldA_nc(WITIm, PADK, mbase, k0);
    v2f br = ldB_slab(inre, k0, nbase);
    v2f bi = ldB_slab(inim, k0, nbase);
    v2f af = ar + ai;
    v2f bf = br + bi;
    p1 = wmma4(ar, br, p1);
    p2 = wmma4(ai, bi, p2);
    p3 = wmma4(af, bf, p3);
  }
  int lane = threadIdx.x & 31;
  int n = nbase + (lane & 15);
  int mo = mbase + 8 * (lane >> 4);
  if (n < TL2) {
#pragma unroll
    for (int r = 0; r < 8; ++r) {
      int m = mo + r;
      if (m < TL1) {
        int idx = (slab * TL1 + m) * TL2 + n;
        Vre[idx] = p1[r] - p2[r];
        Vim[idx] = p3[r] - p1[r] - p2[r];
      }
    }
  }
}

// ---------------------------------------------------------------------------
// Kernel 8: pole-exponential branch, writes x2/(L1*L2) into d_out.
// ---------------------------------------------------------------------------
__global__ __launch_bounds__(256) void k_x2(const float* res2re,
                                            const float* res2im,
                                            const float* e1re,
                                            const float* e1im,
                                            const float* e2re,
                                            const float* e2im, float* out) {
  int blk = blockIdx.x;
  int b = blk / CO, i2 = blk % CO;
  int tid = threadIdx.x;
  __shared__ float s_r2t[M1 * M2 * 2];
  __shared__ float s_U[M1 * TL2 * 2];
  __shared__ float s_e1[M1 * TL1 * 2];
  const int NOUT = TL1 * TL2;
  float acc[7];
#pragma unroll
  for (int j = 0; j < 7; ++j) acc[j] = 0.f;
  for (int c = 0; c < CO; ++c) {
    int ci_idx = c * CO + i2;
    for (int t = tid; t < M1 * M2; t += blockDim.x) {
      s_r2t[2 * t] = res2re[(b * CO + c) * M1 * M2 + t];
      s_r2t[2 * t + 1] = res2im[(b * CO + c) * M1 * M2 + t];
    }
    for (int t = tid; t < M1 * TL1; t += blockDim.x) {
      s_e1[2 * t] = e1re[ci_idx * M1 * TL1 + t];
      s_e1[2 * t + 1] = e1im[ci_idx * M1 * TL1 + t];
    }
    __syncthreads();
    // U[p][z] = sum_q res2[p][q] * e2[q][z]
    for (int t = tid; t < M1 * TL2; t += blockDim.x) {
      int p = t / TL2, z = t % TL2;
      float ur = 0.f, ui = 0.f;
      for (int q = 0; q < M2; ++q) {
        float rr = s_r2t[(p * M2 + q) * 2], ri = s_r2t[(p * M2 + q) * 2 + 1];
        float er = e2re[(ci_idx * M2 + q) * TL2 + z];
        float ei = e2im[(ci_idx * M2 + q) * TL2 + z];
        ur += rr * er - ri * ei;
        ui += rr * ei + ri * er;
      }
      s_U[2 * t] = ur; s_U[2 * t + 1] = ui;
    }
    __syncthreads();
    int j = 0;
    for (int t = tid; t < NOUT; t += blockDim.x, ++j) {
      int tt = t / TL2, z = t % TL2;
      float v = 0.f;
      for (int p = 0; p < M1; ++p) {
        float er = s_e1[(p * TL1 + tt) * 2], ei = s_e1[(p * TL1 + tt) * 2 + 1];
        float ur = s_U[(p * TL2 + z) * 2], ui = s_U[(p * TL2 + z) * 2 + 1];
        v += er * ur - ei * ui;  // Re(e1 * U)
      }
      acc[j] += v;
    }
    __syncthreads();
  }
  const float scale = 1.0f / (float)(TL1 * TL2);
  int j = 0;
  for (int t = tid; t < NOUT; t += blockDim.x, ++j)
    out[blk * NOUT + t] = acc[j] * scale;
}

// ---------------------------------------------------------------------------
// Kernel 9: inverse z-DFT, real part only, scaled, accumulated into d_out:
// out[m][z] += (1/(L1*L2)) * (Vre*cos - Vim*sin).
// ---------------------------------------------------------------------------
__global__ __launch_bounds__(256) void k_izdft(const float* Vre,
                                               const float* Vim,
                                               const float* WIZC,
                                               const float* WIZS, float* out) {
  int wave = blockIdx.x * (blockDim.x >> 5) + (threadIdx.x >> 5);
  int mt = wave / 3, nt = wave % 3;
  int mbase = mt * 16, nbase = nt * 16;
  v8f pc = {}; v8f ps = {};
#pragma unroll
  for (int k0 = 0; k0 < TL2; k0 += 4) {
    v2f arr = ldA_nc(Vre, TL2, mbase, k0);
    v2f aii = ldA_nc(Vim, TL2, mbase, k0);
    v2f bc = ldB_nc(WIZC, PADK, k0, nbase);
    v2f bs = ldB_nc(WIZS, PADK, k0, nbase);
    pc = wmma4(arr, bc, pc);
    ps = wmma4(aii, bs, ps);
  }
  const float scale = 1.0f / (float)(TL1 * TL2);
  int lane = threadIdx.x & 31;
  int n = nbase + (lane & 15);
  int mo = mbase + 8 * (lane >> 4);
  if (n < TL2) {
#pragma unroll
    for (int r = 0; r < 8; ++r) {
      int idx = (mo + r) * TL2 + n;
      out[idx] += scale * (pc[r] - ps[r]);
    }
  }
}

// ---------------------------------------------------------------------------
// Host orchestration.
// ---------------------------------------------------------------------------
extern "C" void kernel_launch(void* const* d_in, const int* in_sizes, int n_in,
                              void* d_out, int out_size, void* d_ws,
                              size_t ws_size, hipStream_t stream) {
  (void)in_sizes; (void)n_in; (void)out_size; (void)ws_size;
  const float* X  = (const float*)d_in[0];  // (64,32,41,40) f32
  const float* W1 = (const float*)d_in[1];  // (32,32,8)  complex64 interleaved
  const float* W2 = (const float*)d_in[2];  // (32,32,8)  complex64
  const float* WR = (const float*)d_in[3];  // (32,32,8,8) complex64
  float* out = (float*)d_out;

  float* w = (float*)d_ws;
  size_t off = 0;
  auto alloc = [&](size_t n) { float* p = w + off; off += n; return p; };

  const size_t MAT = (size_t)PADK * PADK;           // 2304
  const size_t NA  = (size_t)NB * CI * TL1 * TL2;   // 3,358,720
  float* WZC  = alloc(MAT); float* WZS  = alloc(MAT);
  float* WTRe = alloc(MAT); float* WTIm = alloc(MAT);
  float* WITRe = alloc(MAT); float* WITIm = alloc(MAT);
  float* WIZC = alloc(MAT); float* WIZS = alloc(MAT);
  float* r1re = alloc((size_t)CI * CO * TL1 * M1);
  float* r1im = alloc((size_t)CI * CO * TL1 * M1);
  float* r2re = alloc((size_t)CI * CO * TL2 * M2);
  float* r2im = alloc((size_t)CI * CO * TL2 * M2);
  float* e1re = alloc((size_t)CI * CO * M1 * TL1);
  float* e1im = alloc((size_t)CI * CO * M1 * TL1);
  float* e2re = alloc((size_t)CI * CO * M2 * TL2);
  float* e2im = alloc((size_t)CI * CO * M2 * TL2);
  float* HfRe = alloc((size_t)TL1 * TL2 * CI * CO);
  float* HfIm = alloc((size_t)TL1 * TL2 * CI * CO);
  float* aZre = alloc(NA); float* aZim = alloc(NA);
  float* aRe  = alloc(NA); float* aIm  = alloc(NA);
  float* aFre = alloc(NA); float* aFim = alloc(NA);
  float* res2re = alloc((size_t)NB * CO * M1 * M2);
  float* res2im = alloc((size_t)NB * CO * M1 * M2);
  // buffer reuse (stream-ordered lifetimes):
  float* R1re = aRe;  float* R1im = aIm;   // alpha dead after k_S
  float* Vre  = aZre; float* Vim  = aZim;  // alphaZ dead after k_tdft

  k_const<<<1, 256, 0, stream>>>(WZC, WZS, WTRe, WTIm, WITRe, WITIm, WIZC,
                                 WIZS);
  k_weights<<<CI * CO, 256, 0, stream>>>(W1, W2, WR, r1re, r1im, r2re, r2im,
                                         e1re, e1im, e2re, e2im, HfRe, HfIm);
  // fwd z-DFT: 5248 M-tiles * 3 N-tiles = 15744 waves / 8 = 1968 blocks
  k_zdft<<<1968, 256, 0, stream>>>(X, WZC, WZS, aZre, aZim);
  // 2048 slabs * 9 tiles = 18432 waves / 8 = 2304 blocks
  k_tdft<<<2304, 256, 0, stream>>>(aZre, aZim, WTRe, WTIm, aRe, aIm, aFre,
                                   aFim);
  k_S<<<NB * CO, 256, 0, stream>>>(aRe, aIm, r1re, r1im, r2re, r2im, WR,
                                   res2re, res2im);
  k_mix<<<TL1 * TL2, 256, 0, stream>>>(aFre, aFim, HfRe, HfIm, R1re, R1im);
  k_itdft<<<2304, 256, 0, stream>>>(R1re, R1im, WITRe, WITIm, Vre, Vim);
  k_x2<<<NB * CO, 256, 0, stream>>>(res2re, res2im, e1re, e1im, e2re, e2im,
                                    out);
  k_izdft<<<1968, 256, 0, stream>>>(Vre, Vim, WIZC, WIZS, out);
}